// Decoder_58995670778178
// MI455X (gfx1250) — compile-verified
//
#include <hip/hip_runtime.h>
#include <stdint.h>

#define NB     8        // batch
#define NA     120000   // anchors
#define NC     80       // classes
#define TOPN   1000
#define CAP    2048     // sort buffer (pow2 >= worst-case ties + TOPN)
#define MAXDET 100
#define MIN_SCORE 0.05f
#define NMS_THR   0.5f

typedef __attribute__((ext_vector_type(4))) unsigned int u32x4;
typedef __attribute__((ext_vector_type(8))) int          i32x8;
typedef __attribute__((ext_vector_type(4))) int          i32x4;

// order-preserving float -> uint key (handles sign, monotonic)
__device__ __forceinline__ unsigned sortkey(float f) {
  unsigned u = __float_as_uint(f);
  return u ^ ((unsigned)((int)u >> 31) | 0x80000000u);
}

// ---------------- Kernel 1: per-anchor max/argmax over classes, TDM-staged ----
#define K1_ROWS 128
__global__ __launch_bounds__(K1_ROWS)
void k_argmax(const float* __restrict__ cls,
              float* __restrict__ scores0,
              unsigned* __restrict__ classes0) {
  extern __shared__ float smem[];
  float*    tile = smem;                               // K1_ROWS*NC floats (40 KB)
  float*    resS = smem + K1_ROWS * NC;                // K1_ROWS floats
  unsigned* resC = (unsigned*)(resS + K1_ROWS);        // K1_ROWS uints
  const int img = blockIdx.y;
  const long long row0 = (long long)blockIdx.x * K1_ROWS;
  const unsigned nrem = (unsigned)((long long)NA - row0); // valid rows in tile

  if (threadIdx.x < 32) {                       // wave 0 issues the DMA
    unsigned long long gaddr = (unsigned long long)(uintptr_t)cls +
        (((unsigned long long)img * NA + (unsigned long long)row0) *
         (unsigned long long)NC) * 4ull;
    u32x4 g0; i32x8 g1; i32x4 g2; i32x4 g3; i32x8 g4;
    // ---- D# group 0: count=1 | lds_addr | global_addr[56:0] | type=2
    g0[0] = 1u;
    g0[1] = 0u;                                           // LDS offset 0 (= tile)
    g0[2] = (unsigned)(gaddr & 0xFFFFFFFFull);
    g0[3] = (unsigned)((gaddr >> 32) & 0x1FFFFFFull) | 0x80000000u; // type=2<<30
    // ---- D# group 1: data_size=4B; tensor 80 x nrem; tile 80 x 128; stride0=80
    g1[0] = (int)(2u << 16);                              // data_size code 2 (4B)
    g1[1] = (int)((unsigned)NC << 16);                    // tensor_dim0 lo16
    g1[2] = (int)((nrem & 0xFFFFu) << 16);                // dim0 hi16 | dim1 lo16
    g1[3] = (int)((nrem >> 16) | ((unsigned)NC << 16));   // dim1 hi16 | tile_dim0
    g1[4] = (int)K1_ROWS;                                 // tile_dim1 | tile_dim2=0
    g1[5] = (int)NC;                                      // tensor_dim0_stride lo32
    g1[6] = 0; g1[7] = 0;                                 // stride0 hi / stride1
    g2[0]=g2[1]=g2[2]=g2[3]=0;                            // no dims 2..4
    g3[0]=g3[1]=g3[2]=g3[3]=0;
    g4[0]=g4[1]=g4[2]=g4[3]=g4[4]=g4[5]=g4[6]=g4[7]=0;    // unused extra group
    __builtin_amdgcn_tensor_load_to_lds(g0, g1, g2, g3, g4, 0);
    __builtin_amdgcn_s_wait_tensorcnt(0);
  }
  __syncthreads();

  // wave-cooperative row reduction: conflict-free LDS reads + shfl_xor argmax
  const int lane = threadIdx.x & 31;
  const int wv   = threadIdx.x >> 5;              // 4 waves, 32 rows each
  for (int r = wv * 32; r < wv * 32 + 32; ++r) {
    const float* row = tile + r * NC;
    float m = row[lane];      int c = lane;       // classes 0..31
    float v = row[32 + lane];                     // classes 32..63
    if (v > m) { m = v; c = 32 + lane; }
    if (lane < 16) {                              // classes 64..79
      float v2 = row[64 + lane];
      if (v2 > m) { m = v2; c = 64 + lane; }
    }
    for (int off = 16; off; off >>= 1) {          // max with lowest-class tie
      float om = __shfl_xor(m, off, 32);
      int   oc = __shfl_xor(c, off, 32);
      if (om > m || (om == m && oc < c)) { m = om; c = oc; }
    }
    if (lane == 0) { resS[r] = m; resC[r] = (unsigned)c; }
  }
  __syncthreads();

  long long anchor = row0 + threadIdx.x;          // coalesced stores
  if (anchor < (long long)NA) {
    long long o = (long long)img * NA + anchor;
    scores0[o]  = resS[threadIdx.x];
    classes0[o] = resC[threadIdx.x];
  }
}

// ---------------- Kernel 2: exact top-1000 per image (radix-select + bitonic) --
__global__ __launch_bounds__(1024)
void k_select(const float* __restrict__ scores0,
              unsigned long long* __restrict__ topc) {
  __shared__ unsigned hist[256];
  __shared__ unsigned sh_prefix, sh_need, sh_cnt;
  __shared__ unsigned long long sortbuf[CAP];   // 16 KB
  const int img = blockIdx.x, tid = threadIdx.x;
  const float* sc = scores0 + (long long)img * NA;

  if (tid == 0) { sh_prefix = 0u; sh_need = TOPN; }
  // MSB-first radix select: after 4 passes sh_prefix == K (TOPN-th largest key)
  for (int pass = 0; pass < 4; ++pass) {
    const int shift = 24 - 8 * pass;
    const unsigned maskHigh = pass ? (0xFFFFFFFFu << (shift + 8)) : 0u;
    if (tid < 256) hist[tid] = 0u;
    __syncthreads();
    const unsigned prefix = sh_prefix;
    for (int i = tid; i < NA; i += 1024) {
      __builtin_prefetch(sc + i + 8192, 0, 0);  // global_prefetch_b8 (L2-resident)
      unsigned k = sortkey(sc[i]);
      if ((k & maskHigh) == prefix)
        atomicAdd(&hist[(k >> shift) & 255u], 1u);
    }
    __syncthreads();
    if (tid == 0) {
      unsigned cum = 0, need = sh_need;
      for (int d = 255; d >= 0; --d) {
        unsigned h = hist[d];
        cum += h;
        if (cum >= need) {
          sh_need   = need - (cum - h);
          sh_prefix = prefix | ((unsigned)d << shift);
          break;
        }
      }
    }
    __syncthreads();
  }
  const unsigned K = sh_prefix;

  // gather every key >= K as (key<<32)|~idx ; pad with 0 (sorts last)
  for (int i = tid; i < CAP; i += 1024) sortbuf[i] = 0ull;
  if (tid == 0) sh_cnt = 0u;
  __syncthreads();
  for (int i = tid; i < NA; i += 1024) {
    unsigned k = sortkey(sc[i]);
    if (k >= K) {
      unsigned p = atomicAdd(&sh_cnt, 1u);
      if (p < CAP)
        sortbuf[p] = ((unsigned long long)k << 32) | (unsigned)(~(unsigned)i);
    }
  }
  __syncthreads();

  // bitonic sort, descending (key desc, then index asc via ~idx)
  for (unsigned kk = 2; kk <= CAP; kk <<= 1) {
    for (unsigned j = kk >> 1; j > 0; j >>= 1) {
      for (unsigned i = tid; i < CAP; i += 1024) {
        unsigned ixj = i ^ j;
        if (ixj > i) {
          unsigned long long a = sortbuf[i], b = sortbuf[ixj];
          bool descBlock = ((i & kk) == 0);
          if ((a < b) == descBlock) { sortbuf[i] = b; sortbuf[ixj] = a; }
        }
      }
      __syncthreads();
    }
  }
  for (int i = tid; i < TOPN; i += 1024)
    topc[(long long)img * TOPN + i] = sortbuf[i];
}

// ---------------- Kernel 3: gather + decode -----------------------------------
__global__ void k_decode(const unsigned long long* __restrict__ topc,
                         const float* __restrict__ scores0,
                         const unsigned* __restrict__ classes0,
                         const float* __restrict__ reg,
                         const float* __restrict__ anch,
                         float* __restrict__ tscore,
                         float* __restrict__ tclass,
                         float* __restrict__ tbox) {
  int t = blockIdx.x * blockDim.x + threadIdx.x;
  int img = blockIdx.y;
  if (t >= TOPN) return;
  unsigned long long comp = topc[(long long)img * TOPN + t];
  unsigned idx = ~((unsigned)comp);
  long long base = (long long)img * NA + idx;
  float s = scores0[base];
  unsigned cl = classes0[base];
  const float* r = reg  + base * 4;
  const float* a = anch + base * 4;
  float rx = r[0]*0.1f, ry = r[1]*0.1f, rw = r[2]*0.2f, rh = r[3]*0.2f;
  float awx = a[2]-a[0], awy = a[3]-a[1];
  float acx = a[0] + 0.5f*awx, acy = a[1] + 0.5f*awy;
  float pwx = expf(rw)*awx, pwy = expf(rh)*awy;
  float pcx = rx*awx + acx, pcy = ry*awy + acy;
  float x1 = truncf(pcx - 0.5f*pwx), y1 = truncf(pcy - 0.5f*pwy);
  float x2 = truncf(pcx + 0.5f*pwx), y2 = truncf(pcy + 0.5f*pwy);
  x1 = fmaxf(x1, 0.0f);   y1 = fmaxf(y1, 0.0f);
  x2 = fminf(x2, 639.0f); y2 = fminf(y2, 639.0f);
  long long o = (long long)img * TOPN + t;
  tscore[o] = s;
  tclass[o] = (float)cl;
  tbox[o*4+0]=x1; tbox[o*4+1]=y1; tbox[o*4+2]=x2; tbox[o*4+3]=y2;
}

// ---------------- Kernel 4: greedy NMS + compaction ---------------------------
__global__ __launch_bounds__(1024)
void k_nms(const float* __restrict__ tscore, const float* __restrict__ tclass,
           const float* __restrict__ tbox, float* __restrict__ out) {
  __shared__ float bx1[TOPN], by1[TOPN], bx2[TOPN], by2[TOPN], sarea[TOPN];
  __shared__ unsigned char keep[TOPN];
  __shared__ int ranks[TOPN];
  const int img = blockIdx.x, tid = threadIdx.x;
  const float* sc = tscore + (long long)img * TOPN;

  for (int i = tid; i < TOPN; i += 1024) {
    long long o = (long long)(img * TOPN + i) * 4;
    float x1 = tbox[o+0], y1 = tbox[o+1], x2 = tbox[o+2], y2 = tbox[o+3];
    bx1[i]=x1; by1[i]=y1; bx2[i]=x2; by2[i]=y2;
    sarea[i] = (x2 - x1) * (y2 - y1);
    keep[i] = (sc[i] > MIN_SCORE) ? 1 : 0;
  }
  float* os = out + img * MAXDET;
  float* oc = out + NB * MAXDET + img * MAXDET;
  float* ob = out + 2 * NB * MAXDET + img * MAXDET * 4;
  for (int i = tid; i < MAXDET; i += 1024) {
    os[i] = -1.0f; oc[i] = -1.0f;
    ob[i*4+0]=-1.0f; ob[i*4+1]=-1.0f; ob[i*4+2]=-1.0f; ob[i*4+3]=-1.0f;
  }
  __syncthreads();

  for (int i = 0; i < TOPN - 1; ++i) {
    if (keep[i]) {
      float ax1=bx1[i], ay1=by1[i], ax2=bx2[i], ay2=by2[i], aa=sarea[i];
      for (int j = i + 1 + tid; j < TOPN; j += 1024) {
        if (!keep[j]) continue;
        float lx = fmaxf(ax1, bx1[j]), ly = fmaxf(ay1, by1[j]);
        float rx = fminf(ax2, bx2[j]), ry = fminf(ay2, by2[j]);
        float iw = fmaxf(rx - lx, 0.0f), ih = fmaxf(ry - ly, 0.0f);
        float inter = iw * ih;
        float iou = inter / (aa + sarea[j] - inter);
        if (iou > NMS_THR) keep[j] = 0;
      }
    }
    __syncthreads();
  }

  if (tid == 0) {                     // stable compaction ranks
    int r = 0;
    for (int j = 0; j < TOPN; ++j) { ranks[j] = keep[j] ? r : -1; r += keep[j]; }
  }
  __syncthreads();
  for (int j = tid; j < TOPN; j += 1024) {
    int rk = ranks[j];
    if (rk >= 0 && rk < MAXDET) {
      os[rk] = sc[j];
      oc[rk] = tclass[img * TOPN + j];
      ob[rk*4+0]=bx1[j]; ob[rk*4+1]=by1[j]; ob[rk*4+2]=bx2[j]; ob[rk*4+3]=by2[j];
    }
  }
}

// ---------------- Launcher ----------------------------------------------------
extern "C" void kernel_launch(void* const* d_in, const int* in_sizes, int n_in,
                              void* d_out, int out_size, void* d_ws, size_t ws_size,
                              hipStream_t stream) {
  (void)in_sizes; (void)n_in; (void)out_size; (void)ws_size;
  const float* cls  = (const float*)d_in[0];
  const float* reg  = (const float*)d_in[1];
  const float* anch = (const float*)d_in[2];
  char* ws = (char*)d_ws;
  // workspace layout (all 8B-aligned): ~7.94 MB total
  float*              scores0  = (float*)             (ws + 0);        // 3,840,000 B
  unsigned*           classes0 = (unsigned*)          (ws + 3840000);  // 3,840,000 B
  unsigned long long* topc     = (unsigned long long*)(ws + 7680000);  //    64,000 B
  float*              tscore   = (float*)             (ws + 7744000);  //    32,000 B
  float*              tclass   = (float*)             (ws + 7776000);  //    32,000 B
  float*              tbox     = (float*)             (ws + 7808000);  //   128,000 B

  dim3 g1((NA + K1_ROWS - 1) / K1_ROWS, NB);
  size_t smem1 = (size_t)(K1_ROWS * NC + K1_ROWS) * sizeof(float)
               + (size_t)K1_ROWS * sizeof(unsigned);   // tile + resS + resC
  k_argmax<<<g1, K1_ROWS, smem1, stream>>>(cls, scores0, classes0);
  k_select<<<NB, 1024, 0, stream>>>(scores0, topc);
  dim3 g3((TOPN + 255) / 256, NB);
  k_decode<<<g3, 256, 0, stream>>>(topc, scores0, classes0, reg, anch,
                                   tscore, tclass, tbox);
  k_nms<<<NB, 1024, 0, stream>>>(tscore, tclass, tbox, (float*)d_out);
}